// GCNII_42992622633735
// MI455X (gfx1250) — compile-verified
//
#include <hip/hip_runtime.h>
#include <math.h>

// GCNII forward for MI455X (gfx1250, wave32).
// N=50000 nodes, E=800000 edges, IN=512, HID=64, OUT=40, L=16.
#define NNODES  50000
#define NEDGES  800000
#define FIN     512
#define HID     64
#define FOUT    40
#define NLAYERS 16
#define ALPHA_C 0.1f
#define LAMBDA_C 0.5f

typedef float v2f __attribute__((ext_vector_type(2)));
typedef float v8f __attribute__((ext_vector_type(8)));

#if __has_builtin(__builtin_amdgcn_wmma_f32_16x16x4_f32)
#define USE_WMMA 1
#else
#define USE_WMMA 0
#endif

// ---------------------------------------------------------------------------
// Degree accumulation (float counts are exact up to 2^24; max degree << that)
// ---------------------------------------------------------------------------
__global__ void k_degrees(const int* __restrict__ src, const int* __restrict__ dst,
                          float* __restrict__ deg_out, float* __restrict__ deg_in) {
  int i = blockIdx.x * blockDim.x + threadIdx.x;
  if (i < NEDGES) {
    atomicAdd(&deg_out[src[i]], 1.0f);
    atomicAdd(&deg_in[dst[i]], 1.0f);
  }
}

__global__ void k_norms(float* __restrict__ d_out, float* __restrict__ d_in) {
  int i = blockIdx.x * blockDim.x + threadIdx.x;
  if (i < NNODES) {
    d_out[i] = rsqrtf(fmaxf(d_out[i], 1.0f));
    d_in[i]  = rsqrtf(fmaxf(d_in[i], 1.0f));
  }
}

// ---------------------------------------------------------------------------
// Fold the identity-residual into the layer weights:
//   W1e = (1-beta)*I + beta*W1 ;  W2e = (1-beta)*I + beta*W2
// so each layer is exactly rst = f@W1e + f0@W2e + bias.
// ---------------------------------------------------------------------------
__global__ void k_effw(const float* __restrict__ W1, const float* __restrict__ W2,
                       float* __restrict__ W1e, float* __restrict__ W2e) {
  int i = blockIdx.x * blockDim.x + threadIdx.x;
  if (i < NLAYERS * HID * HID) {
    int l = i / (HID * HID);
    int r = i % (HID * HID);
    int row = r / HID, col = r % HID;
    float beta = log1pf(LAMBDA_C / (float)(l + 1));
    float id = (row == col) ? 1.0f : 0.0f;
    W1e[i] = (1.0f - beta) * id + beta * W1[i];
    W2e[i] = (1.0f - beta) * id + beta * W2[i];
  }
}

// ---------------------------------------------------------------------------
// h0 = relu(features @ W_in + b_in)   [50000,512]x[512,64]
// One wave32 per 16x16 output tile; K marched 4 at a time with
// V_WMMA_F32_16X16X4_F32 (fp32 matches reference precision; GEMM is not the
// bottleneck so low precision buys nothing here).
// ---------------------------------------------------------------------------
__global__ __launch_bounds__(256)
void k_gemm_in(const float* __restrict__ F, const float* __restrict__ Win,
               const float* __restrict__ bin, float* __restrict__ h0) {
  const int wid = blockIdx.x * (blockDim.x >> 5) + (threadIdx.x >> 5);
  const int nTiles = (NNODES / 16) * (HID / 16);
  if (wid >= nTiles) return;               // uniform per wave -> EXEC stays full
  const int mt = wid >> 2;                 // HID/16 == 4 column tiles
  const int nt = wid & 3;
  const int lane = threadIdx.x & 31;
  const int half = lane >> 4;              // 0: K 0..1, 1: K 2..3 (A frag)
  const int l15  = lane & 15;
  v8f c = {};
#if USE_WMMA
  const int arow = mt * 16 + l15;
  const float* Fp = F + arow * FIN;
  const float* Bp = Win + nt * 16 + l15;
  for (int k = 0; k < FIN; k += 4) {
    const int ka = k + 2 * half;
    float2 av = *(const float2*)(Fp + ka);
    v2f a; a.x = av.x; a.y = av.y;
    v2f b; b.x = Bp[ka * HID]; b.y = Bp[(ka + 1) * HID];
    c = __builtin_amdgcn_wmma_f32_16x16x4_f32(false, a, false, b, (short)0, c,
                                              false, false);
  }
#else
  for (int k = 0; k < FIN; ++k) {
    const float bv = Win[k * HID + nt * 16 + l15];
#pragma unroll
    for (int v = 0; v < 8; ++v)
      c[v] += F[(mt * 16 + v + 8 * half) * FIN + k] * bv;
  }
#endif
  const float bb = bin[nt * 16 + l15];
#pragma unroll
  for (int v = 0; v < 8; ++v) {
    const int M = mt * 16 + v + 8 * half;
    h0[M * HID + nt * 16 + l15] = fmaxf(c[v] + bb, 0.0f);
  }
}

// ---------------------------------------------------------------------------
// Scatter: agg[dst] += h[src] * norm_src[src].
// h (12.8MB) is L2-resident on a 192MB L2; atomic f32 adds stay in L2.
// One thread per (edge, 4 columns): float4 gather + 4 global_atomic_add_f32.
// ---------------------------------------------------------------------------
__global__ __launch_bounds__(256)
void k_scatter(const float* __restrict__ hin, const float* __restrict__ ns,
               const int* __restrict__ src, const int* __restrict__ dst,
               float* __restrict__ agg) {
  unsigned idx = blockIdx.x * blockDim.x + threadIdx.x;  // E*16 = 12.8M
  unsigned e = idx >> 4;
  if (e >= NEDGES) return;
  int c4 = (idx & 15) << 2;
  int s = src[e], d = dst[e];
  float sc = ns[s];
  float4 v = *(const float4*)(hin + s * HID + c4);
  float* ap = agg + d * HID + c4;
  atomicAdd(ap + 0, v.x * sc);
  atomicAdd(ap + 1, v.y * sc);
  atomicAdd(ap + 2, v.z * sc);
  atomicAdd(ap + 3, v.w * sc);
}

// ---------------------------------------------------------------------------
// Layer combine: h' = relu( f@W1e + f0@W2e + bias ),
//   f  = (1-ALPHA) * norm_dst[:,None] * agg,  f0 = ALPHA * h0.
// One wave per 16x16 tile, two chained WMMA accumulations per K-step.
// ---------------------------------------------------------------------------
__global__ __launch_bounds__(256)
void k_layer(const float* __restrict__ agg, const float* __restrict__ nd,
             const float* __restrict__ h0, const float* __restrict__ W1e,
             const float* __restrict__ W2e, const float* __restrict__ bias,
             float* __restrict__ hout) {
  const int wid = blockIdx.x * (blockDim.x >> 5) + (threadIdx.x >> 5);
  const int nTiles = (NNODES / 16) * (HID / 16);
  if (wid >= nTiles) return;
  const int mt = wid >> 2;
  const int nt = wid & 3;
  const int lane = threadIdx.x & 31;
  const int half = lane >> 4;
  const int l15  = lane & 15;
  v8f c = {};
#if USE_WMMA
  const int arow = mt * 16 + l15;
  const float s1 = (1.0f - ALPHA_C) * nd[arow];
  const float* ap = agg + arow * HID;
  const float* hp = h0 + arow * HID;
  const float* B1 = W1e + nt * 16 + l15;
  const float* B2 = W2e + nt * 16 + l15;
  for (int k = 0; k < HID; k += 4) {
    const int ka = k + 2 * half;
    float2 a1v = *(const float2*)(ap + ka);
    float2 a2v = *(const float2*)(hp + ka);
    v2f a1; a1.x = a1v.x * s1; a1.y = a1v.y * s1;
    v2f b1; b1.x = B1[ka * HID]; b1.y = B1[(ka + 1) * HID];
    c = __builtin_amdgcn_wmma_f32_16x16x4_f32(false, a1, false, b1, (short)0, c,
                                              false, false);
    v2f a2; a2.x = a2v.x * ALPHA_C; a2.y = a2v.y * ALPHA_C;
    v2f b2; b2.x = B2[ka * HID]; b2.y = B2[(ka + 1) * HID];
    c = __builtin_amdgcn_wmma_f32_16x16x4_f32(false, a2, false, b2, (short)0, c,
                                              false, false);
  }
#else
  for (int k = 0; k < HID; ++k) {
    const float b1 = W1e[k * HID + nt * 16 + l15];
    const float b2 = W2e[k * HID + nt * 16 + l15];
#pragma unroll
    for (int v = 0; v < 8; ++v) {
      const int M = mt * 16 + v + 8 * half;
      c[v] += agg[M * HID + k] * ((1.0f - ALPHA_C) * nd[M]) * b1 +
              ALPHA_C * h0[M * HID + k] * b2;
    }
  }
#endif
  const float bb = bias[nt * 16 + l15];
#pragma unroll
  for (int v = 0; v < 8; ++v) {
    const int M = mt * 16 + v + 8 * half;
    hout[M * HID + nt * 16 + l15] = fmaxf(c[v] + bb, 0.0f);
  }
}

// ---------------------------------------------------------------------------
// logits = h @ W_out + b_out ; out = log_softmax(logits).
// OUT=40 is not WMMA-friendly and is ~1.5% of FLOPs: one row per thread,
// W_out (10KB) staged in LDS.
// ---------------------------------------------------------------------------
__global__ __launch_bounds__(128)
void k_out(const float* __restrict__ h, const float* __restrict__ Wout,
           const float* __restrict__ bout, float* __restrict__ out) {
  __shared__ float sW[HID * FOUT];
  __shared__ float sb[FOUT];
  for (int i = threadIdx.x; i < HID * FOUT; i += blockDim.x) sW[i] = Wout[i];
  for (int i = threadIdx.x; i < FOUT; i += blockDim.x) sb[i] = bout[i];
  __syncthreads();
  int row = blockIdx.x * blockDim.x + threadIdx.x;
  if (row >= NNODES) return;
  float hr[HID];
  const float4* hp = (const float4*)(h + row * HID);
#pragma unroll
  for (int i = 0; i < HID / 4; ++i) {
    float4 t = hp[i];
    hr[4 * i + 0] = t.x; hr[4 * i + 1] = t.y;
    hr[4 * i + 2] = t.z; hr[4 * i + 3] = t.w;
  }
  float lg[FOUT];
#pragma unroll 4
  for (int o = 0; o < FOUT; ++o) {
    float acc = sb[o];
#pragma unroll
    for (int k = 0; k < HID; ++k) acc += hr[k] * sW[k * FOUT + o];
    lg[o] = acc;
  }
  float mx = lg[0];
#pragma unroll
  for (int o = 1; o < FOUT; ++o) mx = fmaxf(mx, lg[o]);
  float s = 0.0f;
#pragma unroll
  for (int o = 0; o < FOUT; ++o) s += expf(lg[o] - mx);
  float lse = mx + logf(s);
  float* op = out + row * FOUT;
#pragma unroll
  for (int o = 0; o < FOUT; ++o) op[o] = lg[o] - lse;
}

// ---------------------------------------------------------------------------
extern "C" void kernel_launch(void* const* d_in, const int* in_sizes, int n_in,
                              void* d_out, int out_size, void* d_ws, size_t ws_size,
                              hipStream_t stream) {
  const float* features = (const float*)d_in[0];
  const float* W_in  = (const float*)d_in[1];
  const float* b_in  = (const float*)d_in[2];
  const float* W1    = (const float*)d_in[3];
  const float* W2    = (const float*)d_in[4];
  const float* bL    = (const float*)d_in[5];
  const float* W_out = (const float*)d_in[6];
  const float* b_out = (const float*)d_in[7];
  const int*   src   = (const int*)d_in[8];
  const int*   dst   = (const int*)d_in[9];

  // Workspace layout (floats): ~52.1 MB total.
  float* ws = (float*)d_ws;
  float* norm_src = ws;                                  // N (deg_out -> norm)
  float* norm_dst = norm_src + NNODES;                   // N (deg_in  -> norm)
  float* h0  = norm_dst + NNODES;                        // N*HID
  float* hA  = h0 + (size_t)NNODES * HID;                // N*HID
  float* hB  = hA + (size_t)NNODES * HID;                // N*HID
  float* agg = hB + (size_t)NNODES * HID;                // N*HID
  float* W1e = agg + (size_t)NNODES * HID;               // L*HID*HID
  float* W2e = W1e + NLAYERS * HID * HID;                // L*HID*HID

  // Degrees + symmetric norms
  hipMemsetAsync(norm_src, 0, 2 * NNODES * sizeof(float), stream);
  k_degrees<<<(NEDGES + 255) / 256, 256, 0, stream>>>(src, dst, norm_src, norm_dst);
  k_norms<<<(NNODES + 255) / 256, 256, 0, stream>>>(norm_src, norm_dst);

  // Effective (residual-folded) weights, all layers
  k_effw<<<(NLAYERS * HID * HID + 255) / 256, 256, 0, stream>>>(W1, W2, W1e, W2e);

  // h0 = relu(features @ W_in + b_in): 12500 tiles, 8 waves/block
  const int nTiles = (NNODES / 16) * (HID / 16);
  const int gemmBlocks = (nTiles + 7) / 8;
  k_gemm_in<<<gemmBlocks, 256, 0, stream>>>(features, W_in, b_in, h0);

  const float* cur = h0;
  for (int l = 0; l < NLAYERS; ++l) {
    hipMemsetAsync(agg, 0, (size_t)NNODES * HID * sizeof(float), stream);
    k_scatter<<<(NEDGES * 16) / 256, 256, 0, stream>>>(cur, norm_src, src, dst, agg);
    float* nxt = (l & 1) ? hB : hA;
    k_layer<<<gemmBlocks, 256, 0, stream>>>(agg, norm_dst, h0,
                                            W1e + l * HID * HID,
                                            W2e + l * HID * HID,
                                            bL + l * HID, nxt);
    cur = nxt;
  }

  k_out<<<(NNODES + 127) / 128, 128, 0, stream>>>(cur, W_out, b_out, (float*)d_out);
}